// MessagePassingNet_25348896981718
// MI455X (gfx1250) — compile-verified
//
#include <hip/hip_runtime.h>

// MessagePassingNet for MI455X (gfx1250, wave32):
//   Phase 1: fused gather+scatter-add of edge messages (atomic f32, L2-resident)
//   Phase 2: relu(concat(dst_rep, agg) @ W + b) via V_WMMA_F32_16X16X4_F32
// agg lives in d_out (same shape as the final output); GEMM runs in place.

#define D_DIM 128   // feature dim
#define K_DIM 256   // concat dim = 2*D

typedef __attribute__((ext_vector_type(2))) float v2f;
typedef __attribute__((ext_vector_type(8))) float v8f;

// ---------------------------------------------------------------------------
// Phase 1: one wave per edge. 32 lanes x float4 = 128 floats per row.
// src_rep row is a coalesced global_load_b128; accumulation uses the native
// FP32 global atomic (unsafeAtomicAdd -> global_atomic_add_f32), hitting L2.
// ---------------------------------------------------------------------------
__global__ void __launch_bounds__(256) edge_scatter_kernel(
    const float* __restrict__ src_rep,
    const int*   __restrict__ edge_src,
    const int*   __restrict__ edge_dst,
    float*       __restrict__ agg,
    int n_edges)
{
    const int lane    = threadIdx.x & 31;
    const int wave    = (int)((blockIdx.x * blockDim.x + threadIdx.x) >> 5);
    const int n_waves = (int)((gridDim.x * blockDim.x) >> 5);

    for (int e = wave; e < n_edges; e += n_waves) {
        const int s = edge_src[e];
        const int d = edge_dst[e];
        const float4 v =
            *reinterpret_cast<const float4*>(src_rep + (size_t)s * D_DIM + lane * 4);
        float* p = agg + (size_t)d * D_DIM + lane * 4;
        unsafeAtomicAdd(p + 0, v.x);
        unsafeAtomicAdd(p + 1, v.y);
        unsafeAtomicAdd(p + 2, v.z);
        unsafeAtomicAdd(p + 3, v.w);
    }
}

// ---------------------------------------------------------------------------
// Phase 2: per block: 16 output rows x all 128 cols. 8 waves, each wave owns a
// 16-wide N tile. A = concat(dst_rep[m], agg[m]) staged in LDS (padded stride
// 260 -> conflict-free 16-row access). K walked 4 at a time with
// V_WMMA_F32_16X16X4_F32 (64 WMMA per wave). In-place: agg rows are fully read
// into LDS before the barrier; output rows written only after.
// ---------------------------------------------------------------------------
__global__ void __launch_bounds__(256) node_update_gemm_kernel(
    const float* __restrict__ dst_rep,
    const float* __restrict__ W,      // [K_DIM, D_DIM] row-major
    const float* __restrict__ bias,   // [D_DIM]
    float*       __restrict__ out)    // in: agg rows, out: new_rep rows
{
    __shared__ float As[16][K_DIM + 4];   // stride 260: 260 % 64 == 4 -> no bank conflicts

    const int tid = threadIdx.x;          // 0..255
    const int m0  = blockIdx.x * 16;

    // Stage A: thread t loads column t of every row (coalesced 4B loads).
    #pragma unroll
    for (int j = 0; j < 16; ++j) {
        const int col = tid;
        float v;
        if (col < D_DIM) v = dst_rep[(size_t)(m0 + j) * D_DIM + col];
        else             v = out    [(size_t)(m0 + j) * D_DIM + (col - D_DIM)];
        As[j][col] = v;
    }
    __syncthreads();

    const int lane   = tid & 31;
    const int waveId = tid >> 5;          // 0..7
    const int n0     = waveId * 16;

    // 16x4 f32 A fragment: lanes 0-15 -> K = k0+{0,1}; lanes 16-31 -> K = k0+{2,3}
    const int mA   = lane & 15;
    const int kOff = (lane >> 4) * 2;
    const int nB   = lane & 15;

    v8f acc = {};
    #pragma unroll 8
    for (int k0 = 0; k0 < K_DIM; k0 += 4) {
        v2f a, b;
        a.x = As[mA][k0 + kOff];
        a.y = As[mA][k0 + kOff + 1];
        b.x = W[(size_t)(k0 + kOff)     * D_DIM + n0 + nB];
        b.y = W[(size_t)(k0 + kOff + 1) * D_DIM + n0 + nB];
        // 8 args: (neg_a, A, neg_b, B, c_mod, C, reuse_a, reuse_b)
        acc = __builtin_amdgcn_wmma_f32_16x16x4_f32(
            false, a, false, b, (short)0, acc, false, false);
    }

    // C/D layout: VGPR r -> row M = r + 8*(lane>=16); col N = lane&15.
    const int   nCol  = n0 + nB;
    const float bv    = bias[nCol];
    const int   mBase = m0 + (lane >> 4) * 8;
    #pragma unroll
    for (int r = 0; r < 8; ++r) {
        float v = acc[r] + bv;
        v = v > 0.0f ? v : 0.0f;                       // ReLU
        out[(size_t)(mBase + r) * D_DIM + nCol] = v;   // coalesced per half-wave
    }
}

// ---------------------------------------------------------------------------
// Host side
// ---------------------------------------------------------------------------
extern "C" void kernel_launch(void* const* d_in, const int* in_sizes, int n_in,
                              void* d_out, int out_size, void* d_ws, size_t ws_size,
                              hipStream_t stream) {
    const float* src_rep  = (const float*)d_in[0];
    const float* dst_rep  = (const float*)d_in[1];
    const int*   edge_src = (const int*)  d_in[2];
    const int*   edge_dst = (const int*)  d_in[3];
    const float* W        = (const float*)d_in[4];
    const float* bias     = (const float*)d_in[5];
    float*       out      = (float*)d_out;

    const int n_edges = in_sizes[2];
    const int n_dst   = in_sizes[1] / D_DIM;   // 50000

    // agg accumulator lives in d_out; zero it (graph-capturable memset node).
    hipMemsetAsync(d_out, 0, (size_t)out_size * sizeof(float), stream);

    // Phase 1: fused gather + scatter-add.
    edge_scatter_kernel<<<2048, 256, 0, stream>>>(src_rep, edge_src, edge_dst,
                                                  out, n_edges);

    // Phase 2: WMMA GEMM + bias + ReLU, in place over d_out.
    node_update_gemm_kernel<<<n_dst / 16, 256, 0, stream>>>(dst_rep, W, bias, out);
}